// snn_model_36129264894780
// MI455X (gfx1250) — compile-verified
//
#include <hip/hip_runtime.h>

// ---------------------------------------------------------------------------
// SNN restructured as: S0 build -> GEMM(S0@W1) -> LIF scan -> GEMM(S1s@W2)
// -> LIF scan -> GEMM(S2s@W3) -> LIF scan -> output.
// Weights read exactly once (416MB ~= 18us @ 23.3TB/s) instead of once per
// timestep (26.6GB). GEMMs done in f32 WMMA (v_wmma_f32_16x16x4_f32) to match
// reference f32 numerics (spike thresholds are precision-sensitive).
// ---------------------------------------------------------------------------

typedef __attribute__((ext_vector_type(2))) float v2f;
typedef __attribute__((ext_vector_type(8))) float v8f;

#define SNN_TSTEPS 64
#define SNN_THRES  1.0f
#define SNN_LEAK   0.5f   // (2^1 - 1) / 2^1

// ---------------------------------------------------------------------------
// Prep: input-layer LIF (no leak) -> S0[64][4096]; zero row 0 of S1s/S2s and
// of the output (reference row 0 is all-false).
// ---------------------------------------------------------------------------
__global__ __launch_bounds__(256)
void snn_prep(const float* __restrict__ image,
              float* __restrict__ S0,    // 64 x 4096
              float* __restrict__ S1s,   // 64 x 8192 (zero row 0)
              float* __restrict__ S2s,   // 64 x 8192 (zero row 0)
              float* __restrict__ out)   // 65 x 1024 (zero row 0)
{
    int idx = blockIdx.x * 256 + threadIdx.x;
    if (idx < 4096) {
        // Sequential f32 accumulation to match reference rounding exactly.
        float u = image[idx];
        float m = 0.0f;
        for (int t = 0; t < SNN_TSTEPS; ++t) {
            m += u;
            float s = (m >= SNN_THRES) ? 1.0f : 0.0f;
            m -= SNN_THRES * s;               // input layer: no leak
            S0[t * 4096 + idx] = s;
        }
    }
    int j = idx - 4096;
    if (j >= 0 && j < 8192) { S1s[j] = 0.0f; S2s[j] = 0.0f; }
    if (j >= 0 && j < 1024) { out[j] = 0.0f; }
}

// ---------------------------------------------------------------------------
// C[64][N] = S[64][K] @ W[K][N], K split into CHUNKS deterministic partials.
// Block = 128 threads (4 waves). Each wave: one 64-wide N strip x one K chunk,
// all 4 M-tiles (M=64) kept in registers => W streamed exactly once, A reused.
//
// WMMA f32 16x16x4 fragment layouts (ISA 7.12.2, 32-bit A 16x4 / C-D 16x16):
//   A: lane<16: (M=lane, K={k,k+1}), lane>=16: (M=lane-16, K={k+2,k+3})
//   B: mirrored K striping: VGPR0 lanes 0-15 -> row k, 16-31 -> row k+2;
//      VGPR1 lanes 0-15 -> row k+1, 16-31 -> row k+3; N = lane%16
//   C/D: VGPR r: lanes 0-15 -> M=16*mt+r, lanes 16-31 -> M=16*mt+8+r
// ---------------------------------------------------------------------------
template <int K, int N, int CHUNKS>
__global__ __launch_bounds__(128)
void snn_gemm_wmma(const float* __restrict__ S,
                   const float* __restrict__ W,
                   float* __restrict__ Cp)   // CHUNKS x 64 x N
{
    constexpr int KC = K / CHUNKS;
    constexpr int NW = N / 64;

    const int wave = blockIdx.x * 4 + (threadIdx.x >> 5);
    const int lane = threadIdx.x & 31;
    const int half = lane >> 4;     // 0: lanes 0-15, 1: lanes 16-31
    const int lm   = lane & 15;
    const int n64  = wave % NW;
    const int kc   = wave / NW;
    const int nbase = n64 * 64;
    const int k0 = kc * KC;

    v8f acc[4][4] = {};             // [M-tile][N-tile]

    for (int k = k0; k < k0 + KC; k += 4) {
        // A fragments: one per M-tile; 8B-aligned contiguous pair in S.
        v2f af[4];
#pragma unroll
        for (int mt = 0; mt < 4; ++mt) {
            const float* ap = S + (size_t)(mt * 16 + lm) * K + k + 2 * half;
            af[mt] = *(const v2f*)ap;
        }
#pragma unroll
        for (int nt = 0; nt < 4; ++nt) {
            const int n = nbase + nt * 16 + lm;
            const size_t r0 = (size_t)(k + 2 * half) * N + n;
            v2f bf;
            bf.x = W[r0];
            bf.y = W[r0 + N];
#pragma unroll
            for (int mt = 0; mt < 4; ++mt) {
                acc[mt][nt] = __builtin_amdgcn_wmma_f32_16x16x4_f32(
                    /*neg_a=*/false, af[mt],
                    /*neg_b=*/false, bf,
                    /*c_mod=*/(short)0, acc[mt][nt],
                    /*reuse_a=*/false, /*reuse_b=*/false);
            }
        }
    }

    // Store this chunk's partial C (each element written by exactly one lane).
    float* C = Cp + (size_t)kc * 64 * N;
#pragma unroll
    for (int mt = 0; mt < 4; ++mt) {
#pragma unroll
        for (int nt = 0; nt < 4; ++nt) {
            const int n = nbase + nt * 16 + lm;
#pragma unroll
            for (int r = 0; r < 8; ++r) {
                const int row = mt * 16 + half * 8 + r;
                C[(size_t)row * N + n] = acc[mt][nt][r];
            }
        }
    }
}

// ---------------------------------------------------------------------------
// LIF time-scan: one thread per neuron; sums K-chunk partials in fixed order
// (deterministic), runs 64-step membrane recurrence, emits shifted spikes
// (row t+1 = fire(t)) or final output rows 1..64.
// ---------------------------------------------------------------------------
template <int N, int CHUNKS, bool TO_OUT>
__global__ __launch_bounds__(256)
void snn_scan_lif(const float* __restrict__ Cp,   // CHUNKS x 64 x N
                  float* __restrict__ Snext,      // 64 x N shifted spikes
                  float* __restrict__ out)        // 65 x N output
{
    const int j = blockIdx.x * 256 + threadIdx.x;
    if (j >= N) return;
    float m = 0.0f;
    for (int t = 0; t < SNN_TSTEPS; ++t) {
        float inc = 0.0f;
#pragma unroll
        for (int c = 0; c < CHUNKS; ++c)
            inc += Cp[(size_t)c * 64 * N + (size_t)t * N + j];
        m += inc;
        const float fire = (m >= SNN_THRES) ? 1.0f : 0.0f;
        m = (fire > 0.0f) ? (m - SNN_THRES) : (m * SNN_LEAK);
        if (TO_OUT) {
            out[(size_t)(t + 1) * N + j] = fire;
        } else if (t < SNN_TSTEPS - 1) {
            Snext[(size_t)(t + 1) * N + j] = fire;   // row 0 zeroed in prep
        }
    }
}

// ---------------------------------------------------------------------------
// Launch: 7 kernels, dependencies via stream order. Workspace ~21MB:
//   S0 1MB | S1s 2MB | S2s 2MB | Cp 16MB (reused by all three GEMMs)
// ---------------------------------------------------------------------------
extern "C" void kernel_launch(void* const* d_in, const int* in_sizes, int n_in,
                              void* d_out, int out_size, void* d_ws, size_t ws_size,
                              hipStream_t stream) {
    const float* image = (const float*)d_in[0];
    const float* W1    = (const float*)d_in[1];   // 4096 x 8192
    const float* W2    = (const float*)d_in[2];   // 8192 x 8192
    const float* W3    = (const float*)d_in[3];   // 8192 x 1024
    float* out = (float*)d_out;                   // 65 x 1024

    float* S0  = (float*)d_ws;            // 64*4096
    float* S1s = S0  + 64 * 4096;         // 64*8192
    float* S2s = S1s + 64 * 8192;         // 64*8192
    float* Cp  = S2s + 64 * 8192;         // up to 8*64*8192 partial-C region

    snn_prep<<<48, 256, 0, stream>>>(image, S0, S1s, S2s, out);

    // Layer 1: K=4096, N=8192, 4 K-chunks -> 512 waves -> 128 blocks
    snn_gemm_wmma<4096, 8192, 4><<<128, 128, 0, stream>>>(S0, W1, Cp);
    snn_scan_lif<8192, 4, false><<<32, 256, 0, stream>>>(Cp, S1s, nullptr);

    // Layer 2: K=8192, N=8192, 8 K-chunks -> 1024 waves -> 256 blocks
    snn_gemm_wmma<8192, 8192, 8><<<256, 128, 0, stream>>>(S1s, W2, Cp);
    snn_scan_lif<8192, 8, false><<<32, 256, 0, stream>>>(Cp, S2s, nullptr);

    // Layer 3: K=8192, N=1024, 8 K-chunks -> 128 waves -> 32 blocks
    snn_gemm_wmma<8192, 1024, 8><<<32, 128, 0, stream>>>(S2s, W3, Cp);
    snn_scan_lif<1024, 8, true><<<4, 256, 0, stream>>>(Cp, nullptr, out);
}